// MultiplicativeDiffusionLayer_1176821039622
// MI455X (gfx1250) — compile-verified
//
#include <hip/hip_runtime.h>
#include <stdint.h>

#define TILE_EDGES 1024   // edges per tile per block
#define BLOCK      256
#define EPT        4      // edges per thread per tile (16B = one b128 async load)
#define NBUF       3      // async pipeline depth (tiles in flight)

// ---------------- init: zero the log-accumulator ----------------
__global__ void init_kernel(float* __restrict__ acc, int n) {
    int i = blockIdx.x * blockDim.x + threadIdx.x;
    if (i < n) acc[i] = 0.0f;
}

// ---------------- edge kernel: async-LDS streamed scatter ----------------
__global__ void __launch_bounds__(BLOCK)
edge_kernel(const float* __restrict__ x,
            const int*   __restrict__ srcIdx,
            const int*   __restrict__ dstIdx,
            const float* __restrict__ prob,
            const float* __restrict__ ew,
            float*       __restrict__ acc,
            long long E, int numTiles)
{
    __shared__ __align__(16) int   s_src[NBUF][TILE_EDGES];
    __shared__ __align__(16) int   s_dst[NBUF][TILE_EDGES];
    __shared__ __align__(16) float s_prb[NBUF][TILE_EDGES];

    const int t = threadIdx.x;
    const float w = ew[0];

    // LDS byte addresses of this thread's private 16B slice in buffer 0.
    // (generic pointers to LDS carry the LDS offset in the low 32 bits)
    const uint32_t l_src = (uint32_t)(uintptr_t)(&s_src[0][0]) + (uint32_t)t * 16u;
    const uint32_t l_dst = (uint32_t)(uintptr_t)(&s_dst[0][0]) + (uint32_t)t * 16u;
    const uint32_t l_prb = (uint32_t)(uintptr_t)(&s_prb[0][0]) + (uint32_t)t * 16u;
    const uint32_t bufStride = (uint32_t)TILE_EDGES * 4u;

    // Non-temporal streaming loads: edge data is read exactly once and consumed
    // from LDS, so keep it from evicting the hot x/acc lines in L2.
    auto issue = [&](int b, int tile) {
        uint32_t goff = ((uint32_t)tile * (uint32_t)TILE_EDGES + (uint32_t)t * EPT) * 4u;
        uint32_t bo   = (uint32_t)b * bufStride;
        asm volatile("global_load_async_to_lds_b128 %0, %1, %2 th:TH_LOAD_NT"
                     :: "v"(l_src + bo), "v"(goff), "s"(srcIdx) : "memory");
        asm volatile("global_load_async_to_lds_b128 %0, %1, %2 th:TH_LOAD_NT"
                     :: "v"(l_dst + bo), "v"(goff), "s"(dstIdx) : "memory");
        asm volatile("global_load_async_to_lds_b128 %0, %1, %2 th:TH_LOAD_NT"
                     :: "v"(l_prb + bo), "v"(goff), "s"(prob)   : "memory");
    };

    auto process = [&](int b) {
        const int4   s4 = *reinterpret_cast<const int4*>(&s_src[b][t * EPT]);
        const int4   d4 = *reinterpret_cast<const int4*>(&s_dst[b][t * EPT]);
        const float4 p4 = *reinterpret_cast<const float4*>(&s_prb[b][t * EPT]);
        const int   ss[4] = { s4.x, s4.y, s4.z, s4.w };
        const int   dd[4] = { d4.x, d4.y, d4.z, d4.w };
        const float pp[4] = { p4.x, p4.y, p4.z, p4.w };
#pragma unroll
        for (int k = 0; k < 4; ++k) {
            float m = 1.0f - x[ss[k]] * pp[k] * w;   // m in (0,1]
            atomicAdd(&acc[dd[k]], __logf(m));       // global_atomic_add_f32 (L2)
        }
    };

    // --- triple-buffered async pipeline over tiles (grid-stride) ---
    // Each in-flight tile == 3 async-load instructions (per-wave ASYNCcnt),
    // async loads complete in order, so waiting ASYNCcnt <= 3*(inflight-1)
    // guarantees the oldest tile is resident in LDS.
    int nextTile = blockIdx.x;
    int count = 0;
#pragma unroll
    for (int i = 0; i < NBUF; ++i) {
        if (nextTile < numTiles) {
            issue(i, nextTile);
            nextTile += (int)gridDim.x;
            ++count;
        }
    }

    int buf = 0;
    while (count > 0) {
        if (count == 3)      { asm volatile("s_wait_asynccnt 6" ::: "memory"); }
        else if (count == 2) { asm volatile("s_wait_asynccnt 3" ::: "memory"); }
        else                 { asm volatile("s_wait_asynccnt 0" ::: "memory"); }

        process(buf);
        --count;

        if (nextTile < numTiles) {      // refill the buffer we just drained
            issue(buf, nextTile);
            nextTile += (int)gridDim.x;
            ++count;
        }
        buf = (buf == NBUF - 1) ? 0 : buf + 1;
    }

    // remainder edges (E not multiple of TILE_EDGES) — plain path
    long long base = (long long)numTiles * TILE_EDGES;
    for (long long e = base + (long long)blockIdx.x * blockDim.x + t; e < E;
         e += (long long)gridDim.x * blockDim.x) {
        float m = 1.0f - x[srcIdx[e]] * prob[e] * w;
        atomicAdd(&acc[dstIdx[e]], __logf(m));
    }
}

// ---------------- finalize: out = exp(sum of logs) + self_loop * x ----------------
__global__ void finalize_kernel(const float* __restrict__ acc,
                                const float* __restrict__ x,
                                const float* __restrict__ sl,
                                float* __restrict__ out, int n) {
    int i = blockIdx.x * blockDim.x + threadIdx.x;
    if (i < n) out[i] = __expf(acc[i]) + sl[0] * x[i];
}

extern "C" void kernel_launch(void* const* d_in, const int* in_sizes, int n_in,
                              void* d_out, int out_size, void* d_ws, size_t ws_size,
                              hipStream_t stream) {
    const float* x  = (const float*)d_in[0];   // (N,1)
    const int*   ei = (const int*)  d_in[1];   // (2,E)
    const float* ep = (const float*)d_in[2];   // (E,)
    const float* ew = (const float*)d_in[3];   // (1,)
    const float* sl = (const float*)d_in[4];   // (1,)

    const int       N = in_sizes[0];
    const long long E = (long long)in_sizes[2];
    const int* srcIdx = ei;
    const int* dstIdx = ei + E;

    float* acc = (float*)d_ws;      // N floats of log-accumulator scratch
    float* out = (float*)d_out;

    int nBlocksN = (N + BLOCK - 1) / BLOCK;
    init_kernel<<<nBlocksN, BLOCK, 0, stream>>>(acc, N);

    int numTiles = (int)(E / TILE_EDGES);
    int gBlocks  = numTiles > 0 ? (numTiles < 4096 ? numTiles : 4096) : 1;
    edge_kernel<<<gBlocks, BLOCK, 0, stream>>>(x, srcIdx, dstIdx, ep, ew, acc, E, numTiles);

    finalize_kernel<<<nBlocksN, BLOCK, 0, stream>>>(acc, x, sl, out, N);
}